// MultiScaleMemory_31894427140800
// MI455X (gfx1250) — compile-verified
//
#include <hip/hip_runtime.h>

typedef __attribute__((ext_vector_type(2))) float v2f;
typedef __attribute__((ext_vector_type(8))) float v8f;

#define B_ 4
#define S_ 4096
#define N_ 16
#define D_ 64
#define M_ 4
#define SPLIT 4     // s-splits for new_info partials (ws = SPLIT MiB)
#define CHUNKS 8    // s-chunks per (b,n) in output kernel

// ---------------------------------------------------------------------------
// Kernel 1: new_info partials.  partial[split][b][n][kd][vd] = sum over the
// split's s-range of k[b,s,n,kd] * v[b,s,n,vd]   (K^T V via WMMA f32 16x16x4)
// ---------------------------------------------------------------------------
__global__ __launch_bounds__(256) void msm_newinfo_kernel(
    const float* __restrict__ kp, const float* __restrict__ vp,
    float* __restrict__ partials) {
  __shared__ float lds[D_ * D_];

  const int blk   = blockIdx.x;
  const int split = blk % SPLIT;
  const int bn    = blk / SPLIT;
  const int b     = bn / N_;
  const int n     = bn % N_;
  const int tid   = threadIdx.x;
  const int wave  = tid >> 5;
  const int lane  = tid & 31;
  const int mrow  = lane & 15;   // A row / B col / D col within 16-tile
  const int hi    = lane >> 4;   // lane half
  const int khalf = hi * 2;      // K-pair base within K=4 step

  for (int i = tid; i < D_ * D_; i += 256) lds[i] = 0.0f;
  __syncthreads();

  const int sPerBlock = S_ / SPLIT;      // 1024
  const int sPerWave  = sPerBlock / 8;   // 128
  const int s0 = split * sPerBlock + wave * sPerWave;

  const v8f vzero = {0.f, 0.f, 0.f, 0.f, 0.f, 0.f, 0.f, 0.f};
  v8f acc[4][4];
#pragma unroll
  for (int i = 0; i < 4; ++i)
#pragma unroll
    for (int j = 0; j < 4; ++j) acc[i][j] = vzero;

  for (int ss = 0; ss < sPerWave; ss += 4) {
    const int sb = s0 + ss;
    // this lane reads s-rows (sb+khalf) and (sb+khalf+1)
    const long r0 = ((long)b * S_ + (sb + khalf)) * N_ + n;
    const long r1 = r0 + N_;
    v2f af[4], bf[4];
#pragma unroll
    for (int t = 0; t < 4; ++t) {
      const int d = t * 16 + mrow;
      af[t].x = kp[r0 * D_ + d];   // A[M=d][K=khalf]
      af[t].y = kp[r1 * D_ + d];   // A[M=d][K=khalf+1]
      bf[t].x = vp[r0 * D_ + d];   // B[K=khalf][N=d]
      bf[t].y = vp[r1 * D_ + d];   // B[K=khalf+1][N=d]
    }
#pragma unroll
    for (int i = 0; i < 4; ++i)
#pragma unroll
      for (int j = 0; j < 4; ++j)
        acc[i][j] = __builtin_amdgcn_wmma_f32_16x16x4_f32(
            false, af[i], false, bf[j], (short)0, acc[i][j], false, false);
  }

  // cross-wave reduction via LDS float atomics (ds_add_f32)
#pragma unroll
  for (int i = 0; i < 4; ++i)
#pragma unroll
    for (int j = 0; j < 4; ++j)
#pragma unroll
      for (int r = 0; r < 8; ++r) {
        const int kd = i * 16 + r + 8 * hi;
        const int vd = j * 16 + mrow;
        atomicAdd(&lds[kd * D_ + vd], acc[i][j][r]);
      }
  __syncthreads();

  float* dst = partials + (((long)split * B_ + b) * N_ + n) * (D_ * D_);
  for (int i = tid; i < D_ * D_; i += 256) dst[i] = lds[i];
}

// ---------------------------------------------------------------------------
// Kernel 2: combine memory + new_info into Mc (LDS), then out = Q @ Mc.
// ---------------------------------------------------------------------------
__global__ __launch_bounds__(256) void msm_out_kernel(
    const float* __restrict__ q, const float* __restrict__ mem,
    const float* __restrict__ df, const float* __restrict__ gwv,
    const float* __restrict__ partials, float* __restrict__ out) {
  __shared__ float mc[D_ * D_];

  const int blk   = blockIdx.x;
  const int chunk = blk % CHUNKS;
  const int bn    = blk / CHUNKS;
  const int b     = bn / N_;
  const int n     = bn % N_;
  const int tid   = threadIdx.x;
  const int wave  = tid >> 5;
  const int lane  = tid & 31;
  const int mrow  = lane & 15;
  const int hi    = lane >> 4;
  const int khalf = hi * 2;

  // gates = softmax(gw), sig = sigmoid(df); collapse m-dim coefficients.
  float gw0 = gwv[0], gw1 = gwv[1], gw2 = gwv[2], gw3 = gwv[3];
  const float mx = fmaxf(fmaxf(gw0, gw1), fmaxf(gw2, gw3));
  float e[M_];
  e[0] = __expf(gw0 - mx); e[1] = __expf(gw1 - mx);
  e[2] = __expf(gw2 - mx); e[3] = __expf(gw3 - mx);
  const float esum = e[0] + e[1] + e[2] + e[3];
  float amix[M_];
  float cmix = 0.0f;
#pragma unroll
  for (int m = 0; m < M_; ++m) {
    const float gate = e[m] / esum;
    const float sg   = 1.0f / (1.0f + __expf(-df[m]));
    amix[m] = gate * sg;            // weight on old memory
    cmix   += gate * (1.0f - sg);   // weight on new_info
  }

  // Mc[kd][vd] in LDS
  for (int i = tid; i < D_ * D_; i += 256) {
    float ni = 0.0f;
#pragma unroll
    for (int sp = 0; sp < SPLIT; ++sp)
      ni += partials[(((long)sp * B_ + b) * N_ + n) * (D_ * D_) + i];
    float val = cmix * ni;
#pragma unroll
    for (int m = 0; m < M_; ++m)
      val += amix[m] * mem[(((long)b * M_ + m) * N_ + n) * (D_ * D_) + i];
    mc[i] = val;
  }
  __syncthreads();

  const int sPerChunk = S_ / CHUNKS;  // 512
  const v8f vzero = {0.f, 0.f, 0.f, 0.f, 0.f, 0.f, 0.f, 0.f};

#pragma unroll
  for (int iter = 0; iter < sPerChunk / 128; ++iter) {  // 4 row-blocks/wave
    const int s0 = chunk * sPerChunk + iter * 128 + wave * 16;
    v8f acc[4] = {vzero, vzero, vzero, vzero};
#pragma unroll
    for (int kk = 0; kk < 16; ++kk) {  // K = 64 in steps of 4
      const float* qp =
          q + (((long)b * S_ + (s0 + mrow)) * N_ + n) * D_ + kk * 4 + khalf;
      v2f af;
      af.x = qp[0];
      af.y = qp[1];
#pragma unroll
      for (int j = 0; j < 4; ++j) {  // 4 vd tiles
        v2f bf;
        bf.x = mc[(kk * 4 + khalf) * D_ + j * 16 + mrow];
        bf.y = mc[(kk * 4 + khalf + 1) * D_ + j * 16 + mrow];
        acc[j] = __builtin_amdgcn_wmma_f32_16x16x4_f32(
            false, af, false, bf, (short)0, acc[j], false, false);
      }
    }
#pragma unroll
    for (int j = 0; j < 4; ++j)
#pragma unroll
      for (int r = 0; r < 8; ++r) {
        const int srow = s0 + r + 8 * hi;
        out[(((long)b * S_ + srow) * N_ + n) * D_ + j * 16 + mrow] =
            acc[j][r];
      }
  }
}

// ---------------------------------------------------------------------------
extern "C" void kernel_launch(void* const* d_in, const int* in_sizes, int n_in,
                              void* d_out, int out_size, void* d_ws,
                              size_t ws_size, hipStream_t stream) {
  const float* q   = (const float*)d_in[0];
  const float* kp  = (const float*)d_in[1];
  const float* vp  = (const float*)d_in[2];
  const float* mem = (const float*)d_in[3];
  const float* df  = (const float*)d_in[4];
  const float* gw  = (const float*)d_in[5];
  float* out       = (float*)d_out;
  float* partials  = (float*)d_ws;  // SPLIT * B * N * 64 * 64 floats = 4 MiB

  msm_newinfo_kernel<<<B_ * N_ * SPLIT, 256, 0, stream>>>(kp, vp, partials);
  msm_out_kernel<<<B_ * N_ * CHUNKS, 256, 0, stream>>>(q, mem, df, gw,
                                                       partials, out);
}